// PatternAwareLoss_17609365914135
// MI455X (gfx1250) — compile-verified
//
#include <hip/hip_runtime.h>

// PatternAwareLoss for MI455X (gfx1250, wave32).
// B=64, S=8192, L=24. Memory-bound: ~103MB traffic -> ~4.4us at 23.3TB/s.
// Wave-level fp32 reductions via V_WMMA_F32_16X16X4_F32 (exact, layout-robust).

typedef __attribute__((ext_vector_type(2))) float v2f;
typedef __attribute__((ext_vector_type(8))) float v8f;

#define S_LEN 8192
#define L_DIM 24
#define PATTERN_BOOST 1.2f

// Sum a per-lane fp32 value across all 32 lanes of the wave using the fp32
// WMMA unit. A = {v, 0} per lane (each lane's value appears exactly once in
// the 16x4 A matrix, rest zeros), B = all-ones 4x16. Then
//   sum_m D[m][n] = sum of all A entries = wave total.
// C/D layout: lane n<16 holds rows 0..7 of column n in its
// 8 VGPRs; lane n+16 holds rows 8..15. One shfl_xor(16) completes the sum.
// Requires EXEC == all ones (callers guarantee: no divergence, exact grid).
static __device__ __forceinline__ float wave_sum_wmma(float v) {
  v2f a; a[0] = v;    a[1] = 0.0f;
  v2f b; b[0] = 1.0f; b[1] = 1.0f;
  v8f c = {};
  c = __builtin_amdgcn_wmma_f32_16x16x4_f32(
      /*neg_a=*/false, a, /*neg_b=*/false, b,
      /*c_mod=*/(short)0, c, /*reuse_a=*/false, /*reuse_b=*/false);
  float s = c[0] + c[1] + c[2] + c[3] + c[4] + c[5] + c[6] + c[7];
  s += __shfl_xor(s, 16, 32);
  return s; // wave total, valid on every lane
}

// Numerically-stable BCE with logits, elementwise.
// log1p(exp(-|x|)) ~= __logf(1 + __expf(-|x|)): abs error <= e^-|x| only for
// |x| > 17 (never for N(0,1) logits); uses hardware v_exp_f32/v_log_f32.
static __device__ __forceinline__ float bce_elem(float x, int y) {
  float t = __expf(-fabsf(x));
  return fmaxf(x, 0.0f) - x * (float)y + __logf(1.0f + t);
}

__global__ __launch_bounds__(256) void pattern_loss_stage1(
    const float* __restrict__ logits,   // [B*S, 24]
    const int*   __restrict__ labels,   // [B*S, 24]
    const float* __restrict__ amask,    // [B*S]
    float2*      __restrict__ partials) // [gridDim.x] (num, den)
{
  const int tid  = threadIdx.x;
  const int lane = tid & 31;
  const int wid  = tid >> 5;
  const int idx  = blockIdx.x * 256 + tid;   // exact grid: idx < B*S always

  // --- load this position's 24 logits + 24 labels as 128-bit vectors ---
  const float4* lg = (const float4*)(logits + (size_t)idx * L_DIM);
  const int4*   lb = (const int4*)(labels + (size_t)idx * L_DIM);

  float loss = 0.0f;
  int any_label = 0;
#pragma unroll
  for (int i = 0; i < 6; ++i) {
    float4 x4 = lg[i];
    int4   y4 = lb[i];
    any_label |= (y4.x | y4.y | y4.z | y4.w);
    loss += bce_elem(x4.x, y4.x);
    loss += bce_elem(x4.y, y4.y);
    loss += bce_elem(x4.z, y4.z);
    loss += bce_elem(x4.w, y4.w);
  }
  const bool has = (any_label != 0);

  // --- neighbor has_punct: lanes map to consecutive s within one batch row
  // (S % 32 == 0, so a wave never straddles b). Lane k's neighbor is lane k+1
  // via ballot; lane 31 re-reads labels of s+1 (s==S-1 only occurs on lane 31,
  // 8191 % 32 == 31, and then gets no boost). ---
  const unsigned long long bal = __ballot(has);
  const int s = idx & (S_LEN - 1);
  bool next_has = false;
  if (lane < 31) {
    next_has = (bal >> (lane + 1)) & 1ull;
  } else if (s < S_LEN - 1) {
    const int4* nb = (const int4*)(labels + (size_t)(idx + 1) * L_DIM);
    int a2 = 0;
#pragma unroll
    for (int i = 0; i < 6; ++i) {
      int4 y4 = nb[i];
      a2 |= (y4.x | y4.y | y4.z | y4.w);
    }
    next_has = (a2 != 0);
  }

  const float boost = (has && next_has) ? PATTERN_BOOST : 1.0f;
  const float m = amask[idx];

  const float num = loss * boost * m;  // contribution to (loss*mask).sum()
  const float den = m;                 // contribution to mask.sum()

  // --- wave-level exact fp32 reduction on the WMMA unit (EXEC all ones) ---
  const float wnum = wave_sum_wmma(num);
  const float wden = wave_sum_wmma(den);

  __shared__ float snum[8];
  __shared__ float sden[8];
  if (lane == 0) { snum[wid] = wnum; sden[wid] = wden; }
  __syncthreads();
  if (tid == 0) {
    float n = 0.0f, d = 0.0f;
#pragma unroll
    for (int i = 0; i < 8; ++i) { n += snum[i]; d += sden[i]; }
    partials[blockIdx.x] = make_float2(n, d);
  }
}

__global__ __launch_bounds__(256) void pattern_loss_stage2(
    const float2* __restrict__ partials, int nparts, float* __restrict__ out)
{
  const int tid  = threadIdx.x;
  const int lane = tid & 31;
  const int wid  = tid >> 5;

  float n = 0.0f, d = 0.0f;
  for (int i = tid; i < nparts; i += 256) {
    float2 p = partials[i];
    n += p.x;
    d += p.y;
  }
  const float wn = wave_sum_wmma(n);
  const float wd = wave_sum_wmma(d);

  __shared__ float snum[8];
  __shared__ float sden[8];
  if (lane == 0) { snum[wid] = wn; sden[wid] = wd; }
  __syncthreads();
  if (tid == 0) {
    float N = 0.0f, D = 0.0f;
#pragma unroll
    for (int i = 0; i < 8; ++i) { N += snum[i]; D += sden[i]; }
    out[0] = N / D;
  }
}

extern "C" void kernel_launch(void* const* d_in, const int* in_sizes, int n_in,
                              void* d_out, int out_size, void* d_ws, size_t ws_size,
                              hipStream_t stream) {
  const float* logits = (const float*)d_in[0];
  const int*   labels = (const int*)d_in[1];
  const float* amask  = (const float*)d_in[2];
  float*       out    = (float*)d_out;

  const int npos   = in_sizes[2];      // B*S = 524288
  const int blocks = npos / 256;       // 2048 (exact)
  float2* partials = (float2*)d_ws;    // 2048 * 8B = 16KB scratch

  pattern_loss_stage1<<<blocks, 256, 0, stream>>>(logits, labels, amask, partials);
  pattern_loss_stage2<<<1, 256, 0, stream>>>(partials, blocks, out);
}